// S2CHead_1503238553696
// MI455X (gfx1250) — compile-verified
//
#include <hip/hip_runtime.h>
#include <hip/hip_bf16.h>

// Problem constants (fixed by setup_inputs in the reference).
#define V_CNT   2562          // vertices
#define C_CNT   128           // channels
#define NTILES  161           // ceil(2562/16)
#define P_CNT   (NTILES*16)   // 2576, zero-padded vertex count
#define NVOX    (64*64*64)    // 262144 voxels
#define A_FLOATS (NTILES*64)  // 10304 floats = 41216 B, fits easily in 320KB LDS
#define A_VEC4   (A_FLOATS/4) // 2576 16-byte chunks
#define SCALE_F 0.1f
#define EPS_F   1e-8f

typedef __attribute__((ext_vector_type(2))) float v2f;
typedef __attribute__((ext_vector_type(8))) float v8f;

// Monotone float->uint map, truncated to 20 bits, with 12-bit vertex index in
// the low bits: a single u32 compare orders by (quantized score, index).
__device__ __forceinline__ unsigned ordkey(float f, unsigned idx) {
  unsigned u = __float_as_uint(f);
  u = (u & 0x80000000u) ? ~u : (u | 0x80000000u);
  return (u & 0xFFFFF000u) | idx;
}

// Branchless sorted-insert (descending) into 8 register slots; guarded by a
// cheap threshold test so the 16-op max/min chain is skipped once saturated.
__device__ __forceinline__ void insert8(unsigned (&s)[8], unsigned x) {
  if (x > s[7]) {
#pragma unroll
    for (int i = 0; i < 8; ++i) {
      unsigned hi = (s[i] > x) ? s[i] : x;   // v_max_u32
      unsigned lo = (s[i] > x) ? x : s[i];   // v_min_u32
      s[i] = hi;
      x = lo;
    }
  }
}

// CDNA5 async memory->LDS copy (ASYNCcnt tracked, no VGPR round trip).
// vdst = 32-bit LDS byte address, vaddr = 64-bit global address (GV mode).
__device__ __forceinline__ void async_copy_b128(const void* gptr, void* lptr) {
  unsigned lds_addr = (unsigned)(unsigned long long)lptr;  // low 32b == LDS addr
  unsigned long long gaddr = (unsigned long long)gptr;
  asm volatile("global_load_async_to_lds_b128 %0, %1, off"
               :: "v"(lds_addr), "v"(gaddr) : "memory");
}
__device__ __forceinline__ void wait_asynccnt0() {
  asm volatile("s_wait_asynccnt 0x0" ::: "memory");
}

// -------- prep: vertex unit vectors + WMMA-A packed tiles (zero padded) -----
__global__ __launch_bounds__(256)
void prep_vert_kernel(const float* __restrict__ verts,
                      float* __restrict__ a_pack,
                      float* __restrict__ vu) {
  int v = blockIdx.x * blockDim.x + threadIdx.x;
  if (v >= P_CNT) return;
  float ex = 0.f, ey = 0.f, ez = 0.f;
  if (v < V_CNT) {
    float x = verts[v * 3 + 0], y = verts[v * 3 + 1], z = verts[v * 3 + 2];
    float r  = sqrtf(x * x + y * y + z * z);
    float rc = fmaxf(r, 1e-12f);
    ex = x / rc; ey = y / rc; ez = z / rc;   // e_r == xyz / rho
  }
  vu[v * 3 + 0] = ex; vu[v * 3 + 1] = ey; vu[v * 3 + 2] = ez;
  // WMMA 16x4 f32 A layout: lanes 0-15 hold (K0,K1)=(x,y); lanes 16-31 (K2,K3)=(z,0)
  int t = v >> 4, l = v & 15;
  float* tp = a_pack + t * 64;
  tp[l * 2 + 0]      = ex;
  tp[l * 2 + 1]      = ey;
  tp[32 + l * 2 + 0] = ez;
  tp[32 + l * 2 + 1] = 0.f;
}

// -------- prep: d_sph = features @ W_proj.T + b_proj (2562x3, zero padded) ---
__global__ __launch_bounds__(256)
void prep_dsph_kernel(const float* __restrict__ feat,
                      const float* __restrict__ W,
                      const float* __restrict__ b,
                      float* __restrict__ dsph) {
  int tid = blockIdx.x * blockDim.x + threadIdx.x;
  if (tid >= P_CNT * 3) return;
  int v = tid / 3, j = tid % 3;
  float acc = 0.f;
  if (v < V_CNT) {
    acc = b[j];
    const float* f = feat + v * C_CNT;
    const float* w = W + j * C_CNT;
#pragma unroll 4
    for (int k = 0; k < C_CNT; k += 4)
      acc += f[k] * w[k] + f[k + 1] * w[k + 1] + f[k + 2] * w[k + 2] + f[k + 3] * w[k + 3];
  }
  dsph[tid] = acc;
}

// -------- main: WMMA scores + register top-8 + spherical displacement --------
__global__ __launch_bounds__(256)
void voxel_topk_kernel(const float* __restrict__ a_pack,
                       const float* __restrict__ vu,
                       const float* __restrict__ dsph,
                       float* __restrict__ out) {
  // Whole vertex A-operand buffer lives in LDS; it is shared by all 8 waves
  // of this block and re-read 161x per wave (MI455X: 320KB LDS per WGP).
  __shared__ float lds_a[A_FLOATS];

  const int tid  = threadIdx.x;
  const int lane = tid & 31;

  // Async DMA fill: 2576 x 16B, cooperatively across 256 threads.
  for (int i = tid; i < A_VEC4; i += 256)
    async_copy_b128(a_pack + i * 4, lds_a + i * 4);
  wait_asynccnt0();     // drain this wave's async ops
  __syncthreads();      // publish all waves' LDS writes

  const int wave = blockIdx.x * (blockDim.x >> 5) + (tid >> 5);
  const int n    = (wave << 4) + (lane & 15);   // this lane's voxel

  // voxel -> centered cartesian (half-integer grid: rho>0, rxy>0 always)
  const int gx = n & 63, gy = (n >> 6) & 63, gz = n >> 12;
  const float x = (float)gx - 31.5f;
  const float y = (float)gy - 31.5f;
  const float z = (float)gz - 31.5f;
  const float rho  = sqrtf(x * x + y * y + z * z);
  const float irho = 1.0f / rho;
  const float ex = x * irho, ey = y * irho, ez = z * irho;   // e_r

  // WMMA 4x16 f32 B operand: lanes 0-15 hold (K0,K1)=(ex,ey); 16-31 (K2,K3)=(ez,0)
  v2f bop;
  bop.x = (lane < 16) ? ex : ez;
  bop.y = (lane < 16) ? ey : 0.0f;

  unsigned s[8];
#pragma unroll
  for (int i = 0; i < 8; ++i) s[i] = 0u;    // ordkey of any finite float > 0

  const v2f* la   = (const v2f*)lds_a;
  const int mhalf = (lane >> 4) << 3;       // D rows: M=r (lanes 0-15), M=r+8 (16-31)

  // Software-pipelined over tiles: tile t+1's A operand is fetched from LDS
  // while tile t's scores run through the insertion network, so the ds_load
  // latency (s_wait_dscnt) is hidden behind the selection VALU work.
  v2f a = la[lane];                         // tile 0
  for (int t = 0; t < NTILES - 1; ++t) {
    v2f an = la[(t + 1) * 32 + lane];       // prefetch tile t+1
    v8f c = {};
    c = __builtin_amdgcn_wmma_f32_16x16x4_f32(false, a, false, bop,
                                              (short)0, c, false, false);
#pragma unroll
    for (int r = 0; r < 8; ++r) {
      unsigned m = (unsigned)(t * 16 + mhalf + r);   // global vertex id (<4096)
      insert8(s, ordkey(c[r], m));
    }
    a = an;
  }
  { // final tile (peeled)
    v8f c = {};
    c = __builtin_amdgcn_wmma_f32_16x16x4_f32(false, a, false, bop,
                                              (short)0, c, false, false);
#pragma unroll
    for (int r = 0; r < 8; ++r) {
      unsigned m = (unsigned)((NTILES - 1) * 16 + mhalf + r);
      insert8(s, ordkey(c[r], m));
    }
  }

  // Merge the two half-wave partners (disjoint vertex halves of the same voxel).
  unsigned o[8];
#pragma unroll
  for (int r = 0; r < 8; ++r)
    o[r] = (unsigned)__shfl_xor((int)s[r], 16, 32);
#pragma unroll
  for (int r = 0; r < 8; ++r) insert8(s, o[r]);

  if (lane < 16) {
    // Recompute exact f32 dots for the 8 winners -> weights -> d_sph blend.
    float wsum = 0.f, dr = 0.f, dt = 0.f, dp = 0.f;
#pragma unroll
    for (int r = 0; r < 8; ++r) {
      unsigned idx = s[r] & 0xFFFu;
      float vx = vu[idx * 3 + 0], vy = vu[idx * 3 + 1], vz = vu[idx * 3 + 2];
      float d  = ex * vx + ey * vy + ez * vz;
      d = fminf(1.0f, fmaxf(-1.0f, d));
      float w = 1.0f / (1.0f - d + EPS_F);
      wsum += w;
      dr += w * dsph[idx * 3 + 0];
      dt += w * dsph[idx * 3 + 1];
      dp += w * dsph[idx * 3 + 2];
    }
    float iw = 1.0f / wsum;
    dr *= iw; dt *= iw; dp *= iw;

    // spherical basis: e_theta=(ct*cp, ct*sp, -st), e_phi=(-sp, cp, 0)
    const float rxy  = sqrtf(x * x + y * y);
    const float irxy = 1.0f / rxy;
    const float ct = z * irho, st = rxy * irho;
    const float cp = x * irxy, sp = y * irxy;

    const float wt = rho * dt;        // rho * d_theta
    const float wp = rho * st * dp;   // rho * sin(theta) * d_phi

    out[n * 3 + 0] = SCALE_F * (dr * ex + wt * (ct * cp) + wp * (-sp));
    out[n * 3 + 1] = SCALE_F * (dr * ey + wt * (ct * sp) + wp * ( cp));
    out[n * 3 + 2] = SCALE_F * (dr * ez + wt * (-st));
  }
}

extern "C" void kernel_launch(void* const* d_in, const int* in_sizes, int n_in,
                              void* d_out, int out_size, void* d_ws, size_t ws_size,
                              hipStream_t stream) {
  const float* features = (const float*)d_in[0];  // (2562,128)
  const float* verts    = (const float*)d_in[1];  // (2562,3)
  const float* W_proj   = (const float*)d_in[2];  // (3,128)
  const float* b_proj   = (const float*)d_in[3];  // (3,)
  float* out = (float*)d_out;                     // (64,64,64,3)

  float* ws     = (float*)d_ws;
  float* a_pack = ws;                       // NTILES*64      = 10304 floats
  float* vu     = a_pack + NTILES * 64;     // P_CNT*3        =  7728 floats
  float* dsph   = vu + P_CNT * 3;           // P_CNT*3        =  7728 floats

  prep_vert_kernel<<<(P_CNT + 255) / 256, 256, 0, stream>>>(verts, a_pack, vu);
  prep_dsph_kernel<<<(P_CNT * 3 + 255) / 256, 256, 0, stream>>>(features, W_proj,
                                                                b_proj, dsph);
  // 16 voxels per wave, 8 waves per block -> 2048 blocks
  voxel_topk_kernel<<<NVOX / (16 * 8), 256, 0, stream>>>(a_pack, vu, dsph, out);
}